// STAGATE1_16372415332909
// MI455X (gfx1250) — compile-verified
//
#include <hip/hip_runtime.h>
#include <hip/hip_bf16.h>
#include <math.h>

typedef __attribute__((ext_vector_type(2))) float v2f;
typedef __attribute__((ext_vector_type(8))) float v8f;

// ---------------------------------------------------------------------------
// Fast fp32 WMMA GEMM with fragment-packed B and software-pipelined K loop.
//   C[M x Npad] = A[M x K] * B[K x Npad]  (A row-major, B pre-packed)
// Packed B layout: Bp[((k/4)*Npad + c)*4 + (k%4)] = B[k][c]
//   -> lane B fragment = one global_load_b64 at (col*4 + koff).
// Requirements: M%16==0, K%4==0 and K>=4, Npad%(NT*16)==0.
// One wave computes a 16 x (NT*16) strip: NT independent v_wmma per k-step;
// next k-step's fragments are prefetched while current wmma's execute.
// ---------------------------------------------------------------------------
template <int NT>
__global__ void wmma_gemm_f32_fast(const float* __restrict__ A,
                                   const float* __restrict__ Bp,
                                   float* __restrict__ C,
                                   int K, int lda, int ldc,
                                   int Npad, int Nstore,
                                   int tiles_n, int tiles_total) {
    int wave = (int)((blockIdx.x * blockDim.x + threadIdx.x) >> 5);
    int lane = (int)(threadIdx.x & 31);
    if (wave >= tiles_total) return;          // wave-uniform exit
    int tm = wave / tiles_n;
    int tn = wave % tiles_n;

    int row_a = tm * 16 + (lane & 15);
    int koff  = (lane >> 4) * 2;              // lanes 16-31 hold K+2,K+3
    int col0  = tn * (NT * 16) + (lane & 15);

    const float* ap = A + (size_t)row_a * lda + koff;
    const float* bp = Bp + (size_t)col0 * 4 + koff;
    const size_t bstep = (size_t)Npad * 4;    // floats per k-group

    v8f zero = {0.f, 0.f, 0.f, 0.f, 0.f, 0.f, 0.f, 0.f};
    v8f acc[NT];
#pragma unroll
    for (int t = 0; t < NT; ++t) acc[t] = zero;

    // Prologue: fragments for k0 = 0
    v2f a_cur = *(const v2f*)ap;
    v2f b_cur[NT];
#pragma unroll
    for (int t = 0; t < NT; ++t) b_cur[t] = *(const v2f*)(bp + t * 64);

    // Pipelined main loop: prefetch k0+4 while wmma'ing k0
    for (int k0 = 0; k0 < K - 4; k0 += 4) {
        ap += 4;
        bp += bstep;
        v2f a_nxt = *(const v2f*)ap;
        v2f b_nxt[NT];
#pragma unroll
        for (int t = 0; t < NT; ++t) b_nxt[t] = *(const v2f*)(bp + t * 64);
#pragma unroll
        for (int t = 0; t < NT; ++t)
            acc[t] = __builtin_amdgcn_wmma_f32_16x16x4_f32(
                false, a_cur, false, b_cur[t], (short)0, acc[t], false, false);
        a_cur = a_nxt;
#pragma unroll
        for (int t = 0; t < NT; ++t) b_cur[t] = b_nxt[t];
    }
    // Epilogue k-step
#pragma unroll
    for (int t = 0; t < NT; ++t)
        acc[t] = __builtin_amdgcn_wmma_f32_16x16x4_f32(
            false, a_cur, false, b_cur[t], (short)0, acc[t], false, false);

    // C/D layout: VGPR r -> row tm*16 + r (+8 for lanes 16-31)
    int row_c = tm * 16 + (lane >> 4) * 8;
#pragma unroll
    for (int t = 0; t < NT; ++t) {
        int col_c = col0 + t * 16;
        if (col_c < Nstore) {
#pragma unroll
            for (int r = 0; r < 8; ++r)
                C[(size_t)(row_c + r) * ldc + col_c] = acc[t][r];
        }
    }
}

// ---------------------------------------------------------------------------
// Generic B-fragment packer (also does transpose + zero padding in one pass).
//   dst[((k/4)*Np + c)*4 + (k%4)] = valid ? (trans ? src[c*Ksrc+k]
//                                                  : src[k*Nsrc+c]) : 0
// trans=0: src is [Ksrc x Nsrc].  trans=1: src is [Nsrc x Ksrc] (B = src^T).
// ---------------------------------------------------------------------------
__global__ void pack_b(const float* __restrict__ src, float* __restrict__ dst,
                       int Ksrc, int Nsrc, int Kp, int Np, int trans) {
    int i = (int)(blockIdx.x * blockDim.x + threadIdx.x);
    if (i >= Kp * Np) return;
    int j  = i & 3;
    int cg = i >> 2;
    int c  = cg % Np;
    int g  = cg / Np;
    int k  = 4 * g + j;
    float v = 0.f;
    if (k < Ksrc && c < Nsrc)
        v = trans ? src[(size_t)c * Ksrc + k] : src[(size_t)k * Nsrc + c];
    dst[i] = v;
}

// ---------------------------------------------------------------------------
// Per-node attention dot products (wave per node, wave32 shuffle reduce).
// ---------------------------------------------------------------------------
__global__ void att_dots(const float* __restrict__ x1,
                         const float* __restrict__ att_s,
                         const float* __restrict__ att_d,
                         float* __restrict__ as1, float* __restrict__ ad1,
                         int n_nodes, int hid) {
    int wave = (int)((blockIdx.x * blockDim.x + threadIdx.x) >> 5);
    int lane = (int)(threadIdx.x & 31);
    if (wave >= n_nodes) return;
    const float* row = x1 + (size_t)wave * hid;
    float s = 0.f, d = 0.f;
    for (int c = lane; c < hid; c += 32) {
        float v = row[c];
        s += v * att_s[c];
        d += v * att_d[c];
    }
    for (int off = 16; off > 0; off >>= 1) {
        s += __shfl_down(s, off, 32);
        d += __shfl_down(d, off, 32);
    }
    if (lane == 0) { as1[wave] = s; ad1[wave] = d; }
}

// ---------------------------------------------------------------------------
// Edge weights: ew[e] = exp(sigmoid(as1[src]+ad1[dst])); denom[dst] += ew[e].
// sigmoid in (0,1) => segment-max shift unnecessary (softmax shift-invariant).
// ---------------------------------------------------------------------------
__global__ void edge_logits(const int* __restrict__ ei,
                            const float* __restrict__ as1,
                            const float* __restrict__ ad1,
                            float* __restrict__ ew, float* __restrict__ denom,
                            int n_edges) {
    int e = (int)(blockIdx.x * blockDim.x + threadIdx.x);
    if (e >= n_edges) return;
    int s = ei[e];
    int d = ei[n_edges + e];
    float a   = as1[s] + ad1[d];
    float sig = 1.f / (1.f + __expf(-a));
    float x   = __expf(sig);
    ew[e] = x;
    atomicAdd(&denom[d], x);
}

// ---------------------------------------------------------------------------
// Weighted scatter: out[dst,:] += x[src,:] * (ew[e]/(denom[dst]+eps)).
// Wave per edge; f32 global atomics (working set is L2-resident).
// ---------------------------------------------------------------------------
__global__ void edge_scatter(const int* __restrict__ ei,
                             const float* __restrict__ ew,
                             const float* __restrict__ denom,
                             const float* __restrict__ x,
                             float* __restrict__ out,
                             int n_edges, int hid) {
    int wave = (int)((blockIdx.x * blockDim.x + threadIdx.x) >> 5);
    int lane = (int)(threadIdx.x & 31);
    if (wave >= n_edges) return;
    int s = ei[wave];
    int d = ei[n_edges + wave];
    float w = ew[wave] / (denom[d] + 1e-16f);
    const float* xs = x + (size_t)s * hid;
    float* od = out + (size_t)d * hid;
    for (int c = lane; c < hid; c += 32)
        atomicAdd(&od[c], xs[c] * w);
}

__global__ void elu_inplace(float* __restrict__ x, int n) {
    int i = (int)(blockIdx.x * blockDim.x + threadIdx.x);
    if (i >= n) return;
    float v = x[i];
    x[i] = (v > 0.f) ? v : expm1f(v);
}

// h2p [n x src_ld] -> dst [n x ncols] (strip padding columns)
__global__ void copy_cols(const float* __restrict__ src, float* __restrict__ dst,
                          int n, int src_ld, int ncols) {
    int i = (int)(blockIdx.x * blockDim.x + threadIdx.x);
    if (i >= n * ncols) return;
    int r = i / ncols, c = i % ncols;
    dst[i] = src[(size_t)r * src_ld + c];
}

// ---------------------------------------------------------------------------
static inline int ceil_div(int a, int b) { return (a + b - 1) / b; }

static void launch_gemm(const float* A, const float* Bp, float* C,
                        int M, int Npad, int K, int lda, int ldc,
                        int Nstore, hipStream_t stream) {
    int tiles_m = M / 16;                       // M%16==0 guaranteed by caller
    if (Npad % 128 == 0) {
        int tiles_n = Npad / 128;
        int total = tiles_m * tiles_n;
        wmma_gemm_f32_fast<8><<<ceil_div(total, 8), 256, 0, stream>>>(
            A, Bp, C, K, lda, ldc, Npad, Nstore, tiles_n, total);
    } else if (Npad % 32 == 0) {
        int tiles_n = Npad / 32;
        int total = tiles_m * tiles_n;
        wmma_gemm_f32_fast<2><<<ceil_div(total, 8), 256, 0, stream>>>(
            A, Bp, C, K, lda, ldc, Npad, Nstore, tiles_n, total);
    } else {
        int tiles_n = Npad / 16;
        int total = tiles_m * tiles_n;
        wmma_gemm_f32_fast<1><<<ceil_div(total, 8), 256, 0, stream>>>(
            A, Bp, C, K, lda, ldc, Npad, Nstore, tiles_n, total);
    }
}

extern "C" void kernel_launch(void* const* d_in, const int* in_sizes, int n_in,
                              void* d_out, int out_size, void* d_ws, size_t ws_size,
                              hipStream_t stream) {
    const float* features = (const float*)d_in[0];
    const float* W1       = (const float*)d_in[1];
    const float* W2       = (const float*)d_in[2];
    const float* att_src1 = (const float*)d_in[3];
    const float* att_dst1 = (const float*)d_in[4];
    const int*   ei       = (const int*)d_in[5];

    const int HID  = in_sizes[3];               // 512
    const int IN   = in_sizes[1] / HID;         // 1024
    const int OUT  = in_sizes[2] / HID;         // 30
    const int N    = in_sizes[0] / IN;          // 50000 (multiple of 16)
    const int E    = in_sizes[5] / 2;           // 400000
    const int OUTP = ((OUT + 31) / 32) * 32;    // 32 (padded latent dim)

    // Workspace carve-up (floats)
    float* ws    = (float*)d_ws;
    float* x1    = ws;                          // [N, HID]  (reused for x3)
    float* hbuf  = x1 + (size_t)N * HID;        // [N, HID]  (h1 then h3)
    float* h2p   = hbuf + (size_t)N * HID;      // [N, OUTP] padded latent
    float* as1   = h2p + (size_t)N * OUTP;      // [N]
    float* ad1   = as1 + N;                     // [N]
    float* ew    = ad1 + N;                     // [E]
    float* den   = ew + E;                      // [N]
    float* W1p   = den + N;                     // packed W1        [IN  x HID]
    float* W2Pp  = W1p + (size_t)IN * HID;      // packed pad(W2)   [HID x OUTP]
    float* W2Tp  = W2Pp + (size_t)HID * OUTP;   // packed pad(W2^T) [OUTP x HID]
    float* W1Tp  = W2Tp + (size_t)OUTP * HID;   // packed W1^T      [HID x IN]

    float* h2 = (float*)d_out;                  // [N, OUT]
    float* h4 = h2 + (size_t)N * OUT;           // [N, IN]

    // Pack all GEMM B-operands into WMMA fragment order (pad + transpose fused)
    pack_b<<<ceil_div(IN * HID, 256), 256, 0, stream>>>(W1, W1p, IN, HID, IN, HID, 0);
    pack_b<<<ceil_div(HID * OUTP, 256), 256, 0, stream>>>(W2, W2Pp, HID, OUT, HID, OUTP, 0);
    pack_b<<<ceil_div(OUTP * HID, 256), 256, 0, stream>>>(W2, W2Tp, OUT, HID, OUTP, HID, 1);
    pack_b<<<ceil_div(HID * IN, 256), 256, 0, stream>>>(W1, W1Tp, HID, IN, HID, IN, 1);

    // conv1 projection: x1 = features @ W1   [N,1024]x[1024,512]
    launch_gemm(features, W1p, x1, N, HID, IN, IN, HID, HID, stream);

    // node attention scalars
    att_dots<<<ceil_div(N, 8), 256, 0, stream>>>(x1, att_src1, att_dst1, as1, ad1, N, HID);

    // edge softmax weights (computed once; conv3 reuses tied attention)
    hipMemsetAsync(den, 0, (size_t)N * sizeof(float), stream);
    edge_logits<<<ceil_div(E, 256), 256, 0, stream>>>(ei, as1, ad1, ew, den, E);

    // conv1 propagate + ELU: h1 = elu(scatter(x1))
    hipMemsetAsync(hbuf, 0, (size_t)N * HID * sizeof(float), stream);
    edge_scatter<<<ceil_div(E, 8), 256, 0, stream>>>(ei, ew, den, x1, hbuf, E, HID);
    elu_inplace<<<ceil_div(N * HID, 256), 256, 0, stream>>>(hbuf, N * HID);

    // conv2 (linear): h2p = h1 @ pad(W2)  [N,512]x[512,32], strip -> d_out
    launch_gemm(hbuf, W2Pp, h2p, N, OUTP, HID, HID, OUTP, OUTP, stream);
    copy_cols<<<ceil_div(N * OUT, 256), 256, 0, stream>>>(h2p, h2, N, OUTP, OUT);

    // conv3 projection (tied): x3 = h2p @ pad(W2^T)  [N,32]x[32,512]
    launch_gemm(h2p, W2Tp, x1, N, HID, OUTP, OUTP, HID, HID, stream);

    // conv3 propagate + ELU with tied attention weights
    hipMemsetAsync(hbuf, 0, (size_t)N * HID * sizeof(float), stream);
    edge_scatter<<<ceil_div(E, 8), 256, 0, stream>>>(ei, ew, den, x1, hbuf, E, HID);
    elu_inplace<<<ceil_div(N * HID, 256), 256, 0, stream>>>(hbuf, N * HID);

    // conv4 (linear, tied): h4 = h3 @ W1^T  [N,512]x[512,1024]
    launch_gemm(hbuf, W1Tp, h4, N, IN, HID, HID, IN, IN, stream);
}